// GNN_39762807226895
// MI455X (gfx1250) — compile-verified
//
#include <hip/hip_runtime.h>
#include <hip/hip_bf16.h>
#include <math.h>

typedef float v2f __attribute__((ext_vector_type(2)));
typedef float v4f __attribute__((ext_vector_type(4)));
typedef float v8f __attribute__((ext_vector_type(8)));

#define NN 4096          // nodes
#define NE 8192          // edges
#define HD 64            // hidden dim
#define DF 67            // HD + 3
#define HS 72            // H buffer row stride (floats); cols 67..71 are zero pad
#define KE 136           // edge-net L1 K padded (real 134)
#define KM 204           // node-net L1 K padded (real 201), also M stride

__device__ __forceinline__ float sigm(float x) { return 1.0f / (1.0f + expf(-x)); }

// ---------------- decode one-hot incidence -> edge index arrays -------------
__global__ void k_decode(const float* __restrict__ Ri, const float* __restrict__ Ro,
                         int* __restrict__ seg_i, int* __restrict__ seg_o) {
  long gid = (long)blockIdx.x * blockDim.x + threadIdx.x;   // 0 .. 8388607
  const v4f* Ri4 = (const v4f*)Ri;
  const v4f* Ro4 = (const v4f*)Ro;
  v4f a = __builtin_nontemporal_load(&Ri4[gid]);
  v4f b = __builtin_nontemporal_load(&Ro4[gid]);
  int flat = (int)(gid << 2);                               // n*NE + e
#pragma unroll
  for (int c = 0; c < 4; ++c) {
    int f = flat + c;
    if (a[c] != 0.0f) seg_i[f & (NE - 1)] = f >> 13;
    if (b[c] != 0.0f) seg_o[f & (NE - 1)] = f >> 13;
  }
}

// ---------------- deterministic CSR construction ----------------------------
__global__ void k_zero(int* __restrict__ ci, int* __restrict__ co) {
  int g = blockIdx.x * blockDim.x + threadIdx.x;
  if (g < NN) { ci[g] = 0; co[g] = 0; }
}

__global__ void k_hist(const int* __restrict__ si, const int* __restrict__ so,
                       int* __restrict__ ci, int* __restrict__ co) {
  int e = blockIdx.x * blockDim.x + threadIdx.x;
  if (e < NE) { atomicAdd(&ci[si[e]], 1); atomicAdd(&co[so[e]], 1); }
}

__global__ void k_scan(const int* __restrict__ cnt, int* __restrict__ starts,
                       int* __restrict__ cur) {
  __shared__ int sd[1024];
  int t = threadIdx.x;
  int base = t * 4;
  int v0 = cnt[base + 0], v1 = cnt[base + 1], v2 = cnt[base + 2], v3 = cnt[base + 3];
  int local = v0 + v1 + v2 + v3;
  sd[t] = local;
  __syncthreads();
  for (int off = 1; off < 1024; off <<= 1) {
    int x = (t >= off) ? sd[t - off] : 0;
    __syncthreads();
    sd[t] += x;
    __syncthreads();
  }
  int run = sd[t] - local;   // exclusive prefix
  starts[base + 0] = run; cur[base + 0] = run; run += v0;
  starts[base + 1] = run; cur[base + 1] = run; run += v1;
  starts[base + 2] = run; cur[base + 2] = run; run += v2;
  starts[base + 3] = run; cur[base + 3] = run; run += v3;
  if (t == 1023) starts[NN] = run;
}

__global__ void k_place(const int* __restrict__ si, const int* __restrict__ so,
                        int* __restrict__ cur_i, int* __restrict__ cur_o,
                        int* __restrict__ csr_i, int* __restrict__ csr_o) {
  int e = blockIdx.x * blockDim.x + threadIdx.x;
  if (e < NE) {
    int p = atomicAdd(&cur_i[si[e]], 1); csr_i[p] = e;
    int q = atomicAdd(&cur_o[so[e]], 1); csr_o[q] = e;
  }
}

__device__ __forceinline__ void insort(int* a, int p0, int p1) {
  for (int i = p0 + 1; i < p1; ++i) {
    int key = a[i], j = i - 1;
    while (j >= p0 && a[j] > key) { a[j + 1] = a[j]; --j; }
    a[j + 1] = key;
  }
}

__global__ void k_sort(const int* __restrict__ st_i, int* __restrict__ csr_i,
                       const int* __restrict__ st_o, int* __restrict__ csr_o) {
  int n = blockIdx.x * blockDim.x + threadIdx.x;
  if (n < NN) {
    insort(csr_i, st_i[n], st_i[n + 1]);
    insort(csr_o, st_o[n], st_o[n + 1]);
  }
}

// ---------------- input net: H = [tanh(X@Win+bin) | X | 0pad] ---------------
__global__ void k_input(const float* __restrict__ X, const float* __restrict__ Win,
                        const float* __restrict__ bin, float* __restrict__ H) {
  int g = blockIdx.x * blockDim.x + threadIdx.x;
  if (g >= NN * HS) return;
  int n = g / HS;
  int c = g - n * HS;
  if (c < HD) {
    float acc = bin[c];
    acc += X[n * 3 + 0] * Win[0 * HD + c];
    acc += X[n * 3 + 1] * Win[1 * HD + c];
    acc += X[n * 3 + 2] * Win[2 * HD + c];
    H[g] = tanhf(acc);
  } else if (c < DF) {
    H[g] = X[n * 3 + (c - HD)];
  } else {
    H[g] = 0.0f;       // pad cols 67..71 (read by the branchless edge gather)
  }
}

// ---------------- edge net layer 1 (fp32 WMMA 16x16x4) ----------------------
// h_e[e, 0:64] = tanh([H[seg_o[e]] | H[seg_i[e]]] @ We1 + be1)
// Branchless gather: k in [0,67) -> seg_o row, k in [67,136) -> seg_i row,
// where k=134,135 hit the zeroed pad columns of the seg_i row.
__global__ void k_edge_l1(const float* __restrict__ H, const int* __restrict__ si,
                          const int* __restrict__ so, const float* __restrict__ We1,
                          const float* __restrict__ be1, float* __restrict__ h_e) {
  int tid  = threadIdx.x;
  int lane = tid & 31, wv = tid >> 5;
  int idx  = lane & 15, kh = lane >> 4;
  int mt = blockIdx.x;         // 0..511   (16 edges each)
  int nt = wv;                 // 0..3     (16 output cols each)
  int e  = mt * 16 + idx;
  int ro = so[e] * HS;         // base for k < 67
  int ri = si[e] * HS - DF;    // base for k >= 67 (pre-shifted)
  int nc = nt * 16 + idx;
  v8f acc = {0.f, 0.f, 0.f, 0.f, 0.f, 0.f, 0.f, 0.f};
#pragma unroll 4
  for (int s = 0; s < KE / 4; ++s) {
    int k0 = s * 4 + kh * 2;
    int k1 = k0 + 1;
    v2f a, b;
    a.x = H[((k0 < DF) ? ro : ri) + k0];
    a.y = H[((k1 < DF) ? ro : ri) + k1];
    int w0 = (k0 < 133) ? k0 : 133;   // A is zero beyond k=133, clamp W row
    int w1 = (k1 < 133) ? k1 : 133;
    b.x = We1[w0 * HD + nc];
    b.y = We1[w1 * HD + nc];
    acc = __builtin_amdgcn_wmma_f32_16x16x4_f32(false, a, false, b, (short)0, acc,
                                                false, false);
  }
  float bias = be1[nc];
#pragma unroll
  for (int j = 0; j < 8; ++j) {
    int row = mt * 16 + j + 8 * kh;
    h_e[row * HD + nc] = tanhf(acc[j] + bias);
  }
}

// ---------------- edge net layer 2: e_w = sigmoid(h_e @ We2 + be2) ----------
__global__ void k_edge_l2(const float* __restrict__ h_e, const float* __restrict__ We2,
                          const float* __restrict__ be2, float* __restrict__ out) {
  int e = blockIdx.x * blockDim.x + threadIdx.x;
  if (e >= NE) return;
  const v4f* hrow = (const v4f*)(h_e + e * HD);   // 256B-aligned rows
  const v4f* w4   = (const v4f*)We2;
  float acc = be2[0];
#pragma unroll
  for (int j = 0; j < HD / 4; ++j) {
    v4f h = hrow[j];
    v4f w = w4[j];
    acc += h.x * w.x + h.y * w.y + h.z * w.z + h.w * w.w;
  }
  out[e] = sigm(acc);
}

// ---------------- deterministic weighted scatter-aggregate ------------------
// M[n] = [ mi(n) | mo(n) | H[n] | 0pad ]  (stride KM)
__global__ void k_aggregate(const float* __restrict__ H, const float* __restrict__ ew,
                            const int* __restrict__ si, const int* __restrict__ so,
                            const int* __restrict__ st_i, const int* __restrict__ csr_i,
                            const int* __restrict__ st_o, const int* __restrict__ csr_o,
                            float* __restrict__ M) {
  int n = blockIdx.x;
  int t = threadIdx.x;
  if (t < DF) {
    float acc = 0.0f;
    int p0 = st_i[n], p1 = st_i[n + 1];
    for (int p = p0; p < p1; ++p) { int e = csr_i[p]; acc += ew[e] * H[so[e] * HS + t]; }
    M[n * KM + t] = acc;
  } else if (t < 2 * DF) {
    int c = t - DF;
    float acc = 0.0f;
    int p0 = st_o[n], p1 = st_o[n + 1];
    for (int p = p0; p < p1; ++p) { int e = csr_o[p]; acc += ew[e] * H[si[e] * HS + c]; }
    M[n * KM + t] = acc;
  } else if (t < 3 * DF) {
    M[n * KM + t] = H[n * HS + (t - 2 * DF)];
  } else if (t < KM) {
    M[n * KM + t] = 0.0f;
  }
}

// ---------------- node net layer 1 (fp32 WMMA): hN = tanh(M@Wn1+bn1) --------
__global__ void k_node_l1(const float* __restrict__ M, const float* __restrict__ Wn1,
                          const float* __restrict__ bn1, float* __restrict__ hN) {
  int tid  = threadIdx.x;
  int lane = tid & 31, wv = tid >> 5;
  int idx  = lane & 15, kh = lane >> 4;
  int mt = blockIdx.x;          // 0..255
  int nt = wv;
  int nc = nt * 16 + idx;
  const float* arow = M + (mt * 16 + idx) * KM;
  v8f acc = {0.f, 0.f, 0.f, 0.f, 0.f, 0.f, 0.f, 0.f};
#pragma unroll 4
  for (int s = 0; s < KM / 4; ++s) {
    int k0 = s * 4 + kh * 2;
    v2f a = *(const v2f*)(arow + k0);          // 8B-aligned contiguous pair
    int w0 = (k0 < 200) ? k0 : 200;            // A is zero beyond k=200
    int w1 = (k0 + 1 < 200) ? k0 + 1 : 200;
    v2f b;
    b.x = Wn1[w0 * HD + nc];
    b.y = Wn1[w1 * HD + nc];
    acc = __builtin_amdgcn_wmma_f32_16x16x4_f32(false, a, false, b, (short)0, acc,
                                                false, false);
  }
  float bias = bn1[nc];
#pragma unroll
  for (int j = 0; j < 8; ++j) {
    int row = mt * 16 + j + 8 * kh;
    hN[row * HD + nc] = tanhf(acc[j] + bias);
  }
}

// ---------------- node net layer 2 (fp32 WMMA): H[:,0:64]=sigmoid(hN@Wn2+bn2)
__global__ void k_node_l2(const float* __restrict__ hN, const float* __restrict__ Wn2,
                          const float* __restrict__ bn2, float* __restrict__ H) {
  int tid  = threadIdx.x;
  int lane = tid & 31, wv = tid >> 5;
  int idx  = lane & 15, kh = lane >> 4;
  int mt = blockIdx.x;          // 0..255
  int nt = wv;
  int nc = nt * 16 + idx;
  const float* arow = hN + (mt * 16 + idx) * HD;
  v8f acc = {0.f, 0.f, 0.f, 0.f, 0.f, 0.f, 0.f, 0.f};
#pragma unroll
  for (int s = 0; s < HD / 4; ++s) {
    int k0 = s * 4 + kh * 2;
    v2f a = *(const v2f*)(arow + k0);          // 8B-aligned contiguous pair
    v2f b;
    b.x = Wn2[k0 * HD + nc];
    b.y = Wn2[(k0 + 1) * HD + nc];
    acc = __builtin_amdgcn_wmma_f32_16x16x4_f32(false, a, false, b, (short)0, acc,
                                                false, false);
  }
  float bias = bn2[nc];
#pragma unroll
  for (int j = 0; j < 8; ++j) {
    int row = mt * 16 + j + 8 * kh;
    H[row * HS + nc] = sigm(acc[j] + bias);
  }
}

// ---------------------------------------------------------------------------
extern "C" void kernel_launch(void* const* d_in, const int* in_sizes, int n_in,
                              void* d_out, int out_size, void* d_ws, size_t ws_size,
                              hipStream_t stream) {
  (void)in_sizes; (void)n_in; (void)out_size; (void)ws_size;
  const float* X   = (const float*)d_in[0];
  const float* Ri  = (const float*)d_in[1];
  const float* Ro  = (const float*)d_in[2];
  const float* Win = (const float*)d_in[3];
  const float* bin = (const float*)d_in[4];
  const float* We1 = (const float*)d_in[5];
  const float* be1 = (const float*)d_in[6];
  const float* We2 = (const float*)d_in[7];
  const float* be2 = (const float*)d_in[8];
  const float* Wn1 = (const float*)d_in[9];
  const float* bn1 = (const float*)d_in[10];
  const float* Wn2 = (const float*)d_in[11];
  const float* bn2 = (const float*)d_in[12];
  float* out = (float*)d_out;

  char* ws = (char*)d_ws;
  size_t off = 0;
  auto carve = [&](size_t bytes) -> void* {
    void* p = ws + off;
    off += (bytes + 255) & ~(size_t)255;
    return p;
  };
  float* Hbuf  = (float*)carve((size_t)NN * HS * 4);
  float* h_e   = (float*)carve((size_t)NE * HD * 4);
  float* Mbuf  = (float*)carve((size_t)NN * KM * 4);
  float* hN    = (float*)carve((size_t)NN * HD * 4);
  float* e_w   = (float*)carve((size_t)NE * 4);
  int*   seg_i = (int*)carve((size_t)NE * 4);
  int*   seg_o = (int*)carve((size_t)NE * 4);
  int*   cnt_i = (int*)carve((size_t)NN * 4);
  int*   cnt_o = (int*)carve((size_t)NN * 4);
  int*   st_i  = (int*)carve((size_t)(NN + 1) * 4);
  int*   st_o  = (int*)carve((size_t)(NN + 1) * 4);
  int*   cur_i = (int*)carve((size_t)NN * 4);
  int*   cur_o = (int*)carve((size_t)NN * 4);
  int*   csr_i = (int*)carve((size_t)NE * 4);
  int*   csr_o = (int*)carve((size_t)NE * 4);

  // decode one-hot matrices -> per-edge node indices (268 MB stream, NT hint)
  k_decode<<<(NN * (NE / 4)) / 256, 256, 0, stream>>>(Ri, Ro, seg_i, seg_o);

  // deterministic CSR (sorted adjacency) so scatter-adds have fixed order
  k_zero <<<NN / 256, 256, 0, stream>>>(cnt_i, cnt_o);
  k_hist <<<NE / 256, 256, 0, stream>>>(seg_i, seg_o, cnt_i, cnt_o);
  k_scan <<<1, 1024, 0, stream>>>(cnt_i, st_i, cur_i);
  k_scan <<<1, 1024, 0, stream>>>(cnt_o, st_o, cur_o);
  k_place<<<NE / 256, 256, 0, stream>>>(seg_i, seg_o, cur_i, cur_o, csr_i, csr_o);
  k_sort <<<NN / 256, 256, 0, stream>>>(st_i, csr_i, st_o, csr_o);

  // input net
  k_input<<<(NN * HS + 255) / 256, 256, 0, stream>>>(X, Win, bin, Hbuf);

  // 3 message-passing iterations
  for (int it = 0; it < 3; ++it) {
    k_edge_l1 <<<NE / 16, 128, 0, stream>>>(Hbuf, seg_i, seg_o, We1, be1, h_e);
    k_edge_l2 <<<NE / 256, 256, 0, stream>>>(h_e, We2, be2, e_w);
    k_aggregate<<<NN, 256, 0, stream>>>(Hbuf, e_w, seg_i, seg_o,
                                        st_i, csr_i, st_o, csr_o, Mbuf);
    k_node_l1 <<<NN / 16, 128, 0, stream>>>(Mbuf, Wn1, bn1, hN);
    k_node_l2 <<<NN / 16, 128, 0, stream>>>(hN, Wn2, bn2, Hbuf);
  }

  // final edge net -> output [E, 1]
  k_edge_l1<<<NE / 16, 128, 0, stream>>>(Hbuf, seg_i, seg_o, We1, be1, h_e);
  k_edge_l2<<<NE / 256, 256, 0, stream>>>(h_e, We2, be2, out);
}